// InvariantBasedModel_27874337751620
// MI455X (gfx1250) — compile-verified
//
#include <hip/hip_runtime.h>
#include <hip/hip_bf16.h>

// CDNA5 / gfx1250: wave32, WMMA f32 16x16x4 for all four batched GEMMs of the
// ICNN (fwd L0, fwd L1, bwd L1, bwd L0), computed in transposed form so the
// uniform weight matrices are the A operand and per-sample activations are B.
// v2: zero-padded LDS weights/activations (no divergent predicated loads),
//     wave-local LDS fences instead of block barriers (waves run decoupled),
//     16B-aligned staging rows (stride 36) for vectorized DS ops.

typedef float v2f __attribute__((ext_vector_type(2)));
typedef float v8f __attribute__((ext_vector_type(8)));

#define TPB    128
#define NWAVE  4
#define ASTR   36      // staging row stride in floats: 144B, 16B aligned,
                       // 36*lq mod 64 distinct for lq in [0,16) -> no conflicts

__device__ __forceinline__ v8f wmma4(v2f a, v2f b, v8f c) {
  // D = A(16x4 f32) * B(4x16 f32) + C(16x16 f32)
  return __builtin_amdgcn_wmma_f32_16x16x4_f32(false, a, false, b, (short)0, c,
                                               false, false);
}

__device__ __forceinline__ void act_softplus(const v8f& z, v8f& sp, v8f& sg) {
  #pragma unroll
  for (int r = 0; r < 8; ++r) {
    float zz = z[r];
    float e  = __expf(-fabsf(zz));                 // exp(-|z|) <= 1
    sp[r]    = fmaxf(zz, 0.f) + __logf(1.f + e);   // stable softplus
    float rc = 1.f / (1.f + e);
    sg[r]    = (zz >= 0.f) ? rc : e * rc;          // sigmoid(z) = softplus'(z)
  }
}

// Intra-wave LDS exchange fence: LDS ops of a wave execute in order (DScnt),
// so a scheduling fence is sufficient -- no s_barrier needed.
__device__ __forceinline__ void wave_fence() { __builtin_amdgcn_wave_barrier(); }

__global__ __launch_bounds__(TPB)
void icnn_pk_kernel(const float* __restrict__ Fg,
                    const float* __restrict__ W0g, const float* __restrict__ b0g,
                    const float* __restrict__ W1g, const float* __restrict__ b1g,
                    const float* __restrict__ W2g, const float* __restrict__ b2g,
                    float* __restrict__ outP, float* __restrict__ outW)
{
  // W0 zero-padded to 16x32: rows 0..5 = W0[k][j], rows 6..15 = 0.
  // Serves fwd (W0^T access k*32+row, k<8) and bwd (W0[p][k] access, p<16).
  __shared__ __align__(16) float sW0[16 * 32];
  __shared__ __align__(16) float sW1[32 * 32];    // W1[k][j] row-major
  __shared__ __align__(16) float sW2[32];
  __shared__ __align__(16) float sB0[32];
  __shared__ __align__(16) float sB1[32];
  __shared__ __align__(16) float sAct[NWAVE][32][ASTR];  // per-wave [sample][unit]

  const int tid = threadIdx.x;
  for (int i = tid; i < 512;  i += TPB) sW0[i] = (i < 192) ? W0g[i] : 0.f;
  for (int i = tid; i < 1024; i += TPB) sW1[i] = W1g[i];
  if (tid < 32) { sW2[tid] = W2g[tid]; sB0[tid] = b0g[tid]; sB1[tid] = b1g[tid]; }
  __syncthreads();   // only cross-wave dependency in the whole kernel

  const int lane = tid & 31;
  const int half = lane >> 4;      // 0: lanes 0-15, 1: lanes 16-31
  const int lq   = lane & 15;
  const long s   = (long)blockIdx.x * TPB + tid;   // one sample per lane
  float (*act)[ASTR] = sAct[tid >> 5];

  // ---------------- invariants (per lane) ----------------
  const float* Fp = Fg + s * 9;
  float f00=Fp[0], f01=Fp[1], f02=Fp[2];
  float f10=Fp[3], f11=Fp[4], f12=Fp[5];
  float f20=Fp[6], f21=Fp[7], f22=Fp[8];

  float c00 = f00*f00 + f10*f10 + f20*f20;
  float c01 = f00*f01 + f10*f11 + f20*f21;
  float c02 = f00*f02 + f10*f12 + f20*f22;
  float c11 = f01*f01 + f11*f11 + f21*f21;
  float c12 = f01*f02 + f11*f12 + f21*f22;
  float c22 = f02*f02 + f12*f12 + f22*f22;
  float J   = f00*(f11*f22 - f12*f21) - f01*(f10*f22 - f12*f20)
            + f02*(f10*f21 - f11*f20);
  float a00 = c11*c22 - c12*c12;   // diag of cof(C)
  float a11 = c00*c22 - c02*c02;
  float a22 = c00*c11 - c01*c01;
  float I1  = c00 + c11 + c22;

  act[lane][0] = I1;
  act[lane][1] = a00 + a11 + a22;                 // I2 = tr(cof C)
  act[lane][2] = J;
  act[lane][3] = -J;
  act[lane][4] = c00*c00 + c11*c11 + c22*c22;     // I7
  act[lane][5] = a00*a00 + a11*a11 + a22*a22;     // I11
  act[lane][6] = 0.f;                             // K padding for layer 0
  act[lane][7] = 0.f;
  wave_fence();

  // ---------------- layer 0: Z0^T = W0^T (32x8pad) * X^T (8x32) + b0 -------
  v8f acc0[2][2];
  #pragma unroll
  for (int m = 0; m < 2; ++m) {
    v8f bias;
    #pragma unroll
    for (int r = 0; r < 8; ++r) bias[r] = sB0[16*m + r + 8*half];
    acc0[m][0] = bias; acc0[m][1] = bias;
  }
  #pragma unroll
  for (int g = 0; g < 2; ++g) {
    const int k0 = 4*g + 2*half;
    v2f Bt[2];
    #pragma unroll
    for (int n = 0; n < 2; ++n) {
      Bt[n].x = act[16*n + lq][k0];
      Bt[n].y = act[16*n + lq][k0 + 1];
    }
    #pragma unroll
    for (int m = 0; m < 2; ++m) {
      v2f At;
      At.x = sW0[(k0    )*32 + 16*m + lq];   // rows 6,7 are zero-padded
      At.y = sW0[(k0 + 1)*32 + 16*m + lq];
      #pragma unroll
      for (int n = 0; n < 2; ++n) acc0[m][n] = wmma4(At, Bt[n], acc0[m][n]);
    }
  }

  v8f h0, sg0[2][2];
  wave_fence();                 // X fully consumed -> reuse staging buffer
  #pragma unroll
  for (int m = 0; m < 2; ++m)
    #pragma unroll
    for (int n = 0; n < 2; ++n) {
      act_softplus(acc0[m][n], h0, sg0[m][n]);
      #pragma unroll
      for (int r = 0; r < 8; ++r)
        act[16*n + lq][16*m + r + 8*half] = h0[r];   // stage H0^T -> [s][k]
    }
  wave_fence();

  // ---------------- layer 1: Z1^T = W1^T (32x32) * H0^T + b1 ----------------
  v8f acc1[2][2];
  #pragma unroll
  for (int m = 0; m < 2; ++m) {
    v8f bias;
    #pragma unroll
    for (int r = 0; r < 8; ++r) bias[r] = sB1[16*m + r + 8*half];
    acc1[m][0] = bias; acc1[m][1] = bias;
  }
  #pragma unroll
  for (int g = 0; g < 8; ++g) {
    const int k0 = 4*g + 2*half;
    v2f Bt[2];
    #pragma unroll
    for (int n = 0; n < 2; ++n) {
      Bt[n].x = act[16*n + lq][k0];
      Bt[n].y = act[16*n + lq][k0 + 1];
    }
    #pragma unroll
    for (int m = 0; m < 2; ++m) {
      v2f At;
      At.x = sW1[(k0    )*32 + 16*m + lq];   // W1^T[row][k] = W1[k][row]
      At.y = sW1[(k0 + 1)*32 + 16*m + lq];
      #pragma unroll
      for (int n = 0; n < 2; ++n) acc1[m][n] = wmma4(At, Bt[n], acc1[m][n]);
    }
  }

  v8f sg1[2][2];
  wave_fence();
  #pragma unroll
  for (int m = 0; m < 2; ++m)
    #pragma unroll
    for (int n = 0; n < 2; ++n) {
      v8f h1;
      act_softplus(acc1[m][n], h1, sg1[m][n]);
      #pragma unroll
      for (int r = 0; r < 8; ++r)
        act[16*n + lq][16*m + r + 8*half] = h1[r];   // stage H1^T -> [s][k]
    }
  wave_fence();

  // ---------------- output layer (per lane): Wv = H1 @ W2 + b2 --------------
  float z2 = b2g[0];
  #pragma unroll
  for (int k = 0; k < 32; ++k) z2 += sW2[k] * act[lane][k];
  outW[s] = z2;

  // ---------------- backward: dz1 = W2 .* sigmoid(z1) -----------------------
  wave_fence();
  #pragma unroll
  for (int m = 0; m < 2; ++m)
    #pragma unroll
    for (int r = 0; r < 8; ++r) {
      float wr = sW2[16*m + r + 8*half];
      #pragma unroll
      for (int n = 0; n < 2; ++n)
        act[16*n + lq][16*m + r + 8*half] = sg1[m][n][r] * wr;
    }
  wave_fence();

  // dH0^T = W1 (32x32, untransposed) * dZ1^T
  v8f accG[2][2];
  #pragma unroll
  for (int m = 0; m < 2; ++m)
    #pragma unroll
    for (int n = 0; n < 2; ++n)
      #pragma unroll
      for (int r = 0; r < 8; ++r) accG[m][n][r] = 0.f;
  #pragma unroll
  for (int g = 0; g < 8; ++g) {
    const int k0 = 4*g + 2*half;
    v2f Bt[2];
    #pragma unroll
    for (int n = 0; n < 2; ++n) {
      Bt[n].x = act[16*n + lq][k0];
      Bt[n].y = act[16*n + lq][k0 + 1];
    }
    #pragma unroll
    for (int m = 0; m < 2; ++m) {
      v2f At;
      At.x = sW1[(16*m + lq)*32 + k0];       // W1[row][k]
      At.y = sW1[(16*m + lq)*32 + k0 + 1];
      #pragma unroll
      for (int n = 0; n < 2; ++n) accG[m][n] = wmma4(At, Bt[n], accG[m][n]);
    }
  }

  // dz0 = dh0 .* sigmoid(z0); stage
  wave_fence();
  #pragma unroll
  for (int m = 0; m < 2; ++m)
    #pragma unroll
    for (int n = 0; n < 2; ++n)
      #pragma unroll
      for (int r = 0; r < 8; ++r)
        act[16*n + lq][16*m + r + 8*half] = accG[m][n][r] * sg0[m][n][r];
  wave_fence();

  // di^T = W0 (6x32, M zero-padded to 16) * dZ0^T
  v8f accD[2];
  #pragma unroll
  for (int n = 0; n < 2; ++n)
    #pragma unroll
    for (int r = 0; r < 8; ++r) accD[n][r] = 0.f;
  #pragma unroll
  for (int g = 0; g < 8; ++g) {
    const int k0 = 4*g + 2*half;
    v2f At;
    At.x = sW0[lq*32 + k0];                  // rows 6..15 zero-padded
    At.y = sW0[lq*32 + k0 + 1];
    #pragma unroll
    for (int n = 0; n < 2; ++n) {
      v2f Bt;
      Bt.x = act[16*n + lq][k0];
      Bt.y = act[16*n + lq][k0 + 1];
      accD[n] = wmma4(At, Bt, accD[n]);
    }
  }

  wave_fence();
  #pragma unroll
  for (int n = 0; n < 2; ++n)
    #pragma unroll
    for (int r = 0; r < 8; ++r)
      act[16*n + lq][r + 8*half] = accD[n][r];   // di rows 0..5 land in k<6
  wave_fence();

  const float d0 = act[lane][0], d1 = act[lane][1], d2 = act[lane][2];
  const float d3 = act[lane][3], d4 = act[lane][4], d5 = act[lane][5];

  // P = 2 F S + (d2 - d3) cof(F);  S = d0 I + d1 (I1 I - C) + 2 d4 diag(C)
  //                                    + d5 * (full grad of I11 wrt C)
  float s00 = d0 + d1*(I1 - c00) + 2.f*d4*c00 + 2.f*d5*(a11*c22 + a22*c11);
  float s11 = d0 + d1*(I1 - c11) + 2.f*d4*c11 + 2.f*d5*(a00*c22 + a22*c00);
  float s22 = d0 + d1*(I1 - c22) + 2.f*d4*c22 + 2.f*d5*(a00*c11 + a11*c00);
  float s01 = -c01*(d1 + 2.f*d5*a22);
  float s02 = -c02*(d1 + 2.f*d5*a11);
  float s12 = -c12*(d1 + 2.f*d5*a00);
  float dJ  = d2 - d3;

  float g00 = f11*f22 - f12*f21, g01 = f12*f20 - f10*f22, g02 = f10*f21 - f11*f20;
  float g10 = f02*f21 - f01*f22, g11 = f00*f22 - f02*f20, g12 = f01*f20 - f00*f21;
  float g20 = f01*f12 - f02*f11, g21 = f02*f10 - f00*f12, g22 = f00*f11 - f01*f10;

  float* Pp = outP + s * 9;
  Pp[0] = 2.f*(f00*s00 + f01*s01 + f02*s02) + dJ*g00;
  Pp[1] = 2.f*(f00*s01 + f01*s11 + f02*s12) + dJ*g01;
  Pp[2] = 2.f*(f00*s02 + f01*s12 + f02*s22) + dJ*g02;
  Pp[3] = 2.f*(f10*s00 + f11*s01 + f12*s02) + dJ*g10;
  Pp[4] = 2.f*(f10*s01 + f11*s11 + f12*s12) + dJ*g11;
  Pp[5] = 2.f*(f10*s02 + f11*s12 + f12*s22) + dJ*g12;
  Pp[6] = 2.f*(f20*s00 + f21*s01 + f22*s02) + dJ*g20;
  Pp[7] = 2.f*(f20*s01 + f21*s11 + f22*s12) + dJ*g21;
  Pp[8] = 2.f*(f20*s02 + f21*s12 + f22*s22) + dJ*g22;
}

extern "C" void kernel_launch(void* const* d_in, const int* in_sizes, int n_in,
                              void* d_out, int out_size, void* d_ws, size_t ws_size,
                              hipStream_t stream) {
  (void)n_in; (void)out_size; (void)d_ws; (void)ws_size;
  const float* F  = (const float*)d_in[0];
  const float* W0 = (const float*)d_in[1];
  const float* b0 = (const float*)d_in[2];
  const float* W1 = (const float*)d_in[3];
  const float* b1 = (const float*)d_in[4];
  const float* W2 = (const float*)d_in[5];
  const float* b2 = (const float*)d_in[6];
  const int n = in_sizes[0] / 9;          // 2^21, divisible by TPB
  float* outP = (float*)d_out;
  float* outW = outP + (size_t)n * 9;
  const int blocks = n / TPB;
  hipLaunchKernelGGL(icnn_pk_kernel, dim3(blocks), dim3(TPB), 0, stream,
                     F, W0, b0, W1, b1, W2, b2, outP, outW);
}